// CRF_42167988912370
// MI455X (gfx1250) — compile-verified
//
#include <hip/hip_runtime.h>
#include <hip/hip_bf16.h>

// ---------------------------------------------------------------------------
// CRF Viterbi decode for MI455X (gfx1250, wave32).
//   Phase 1: pot = x @ W + b (+ boundary potentials)  -- f32 WMMA 16x16x4
//   Phase 2: per-batch Viterbi forward scan + backtrace, all state in LDS
//            (96KB pot slice + 24KB backpointers per workgroup; uses the
//             CDNA5 320KB-per-WGP LDS).
// Bulk global->LDS staging uses the CDNA5 async-LDS path (ASYNCcnt) when the
// toolchain declares the builtins; otherwise falls back to float4 copies.
// ---------------------------------------------------------------------------

typedef __attribute__((ext_vector_type(2))) float v2f;
typedef __attribute__((ext_vector_type(8))) float v8f;
typedef __attribute__((ext_vector_type(4))) int   v4i;

#define B_SZ   64
#define T_LEN  1024
#define D_DIM  512
#define N_TAGS 24

#if defined(__gfx1250__) && \
    __has_builtin(__builtin_amdgcn_global_load_async_to_lds_b128) && \
    __has_builtin(__builtin_amdgcn_s_wait_asynccnt)
#define HAVE_ASYNC_LDS 1
typedef __attribute__((address_space(1))) v4i* as1_v4i;   // global
typedef __attribute__((address_space(3))) v4i* as3_v4i;   // LDS
#else
#define HAVE_ASYNC_LDS 0
#endif

// Copy `nfloat4` float4s global->LDS, strided by `nthreads` starting at `tid`.
// Caller must barrier afterwards (async path waits its own ASYNCcnt first).
__device__ __forceinline__ void stage_to_lds(float* ldst, const float* gsrc,
                                             int nfloat4, int tid, int nthreads)
{
#if HAVE_ASYNC_LDS
    for (int i = tid; i < nfloat4; i += nthreads) {
        __builtin_amdgcn_global_load_async_to_lds_b128(
            (as1_v4i)(gsrc + 4 * i),
            (as3_v4i)(ldst + 4 * i),
            /*imm offset*/0, /*cpol*/0);
    }
    __builtin_amdgcn_s_wait_asynccnt(0);
#else
    const float4* src = (const float4*)gsrc;
    float4*       dst = (float4*)ldst;
    for (int i = tid; i < nfloat4; i += nthreads)
        dst[i] = src[i];
#endif
}

// ---------------------------------------------------------------------------
// Kernel 1: pot[b*T + t][n] = sum_k x[b,t,k] * W[k,n] + b[n] (+ boundaries)
// One wave per 16-row M tile; two 16-wide N tiles (cols 0..15, 16..23+pad).
// 8 waves (256 threads) per block -> 4096 M tiles / 8 = 512 blocks.
// ---------------------------------------------------------------------------
__global__ __launch_bounds__(256) void crf_pot_gemm(
    const float* __restrict__ x,      // (B*T, D)
    const float* __restrict__ W,      // (D, N)
    const float* __restrict__ bias,   // (N)
    const float* __restrict__ leftb,  // (N)
    const float* __restrict__ rightb, // (N)
    float* __restrict__ pot)          // (B*T, N)
{
    // Stage W in LDS, stride N_TAGS=24: the two half-wave read phases are
    // offset by 2*24=48 banks -> conflict-free across the wave.
    __shared__ __align__(16) float Wlds[D_DIM * N_TAGS];   // 48 KB
    stage_to_lds(Wlds, W, (D_DIM * N_TAGS) / 4, threadIdx.x, 256);
    __syncthreads();

    const int lane  = threadIdx.x & 31;
    const int wave  = threadIdx.x >> 5;
    const int mtile = blockIdx.x * 8 + wave;     // 16-row tile index
    const int row0  = mtile * 16;
    const int hi    = lane >> 4;                 // 0: K+{0,1}, 1: K+{2,3}
    const int lo    = lane & 15;

    const float* xrow = x + (size_t)(row0 + lo) * D_DIM;

    const int  n1     = 16 + lo;                 // second N tile column
    const bool valid1 = (n1 < N_TAGS);
    const int  n1c    = valid1 ? n1 : 0;         // clamp: no EXEC divergence

    v8f c0 = {};
    v8f c1 = {};

    for (int k = 0; k < D_DIM; k += 4) {
        const int ka = k + 2 * hi;
        // A tile (16x4 f32): VGPR0 = A[m][ka], VGPR1 = A[m][ka+1]
        v2f a;
        a.x = xrow[ka];
        a.y = xrow[ka + 1];
        // B tile 0 (4x16): VGPR0 = W[ka][lo], VGPR1 = W[ka+1][lo]
        v2f b0;
        b0.x = Wlds[ka * N_TAGS + lo];
        b0.y = Wlds[(ka + 1) * N_TAGS + lo];
        // B tile 1: columns 16..23 valid, 24..31 forced to zero
        float w0 = Wlds[ka * N_TAGS + n1c];
        float w1 = Wlds[(ka + 1) * N_TAGS + n1c];
        v2f b1;
        b1.x = valid1 ? w0 : 0.0f;
        b1.y = valid1 ? w1 : 0.0f;

        c0 = __builtin_amdgcn_wmma_f32_16x16x4_f32(
                false, a, false, b0, (short)0, c0, false, false);
        c1 = __builtin_amdgcn_wmma_f32_16x16x4_f32(
                false, a, false, b1, (short)0, c1, false, false);
    }

    // C/D layout: VGPR r, lane l -> M = r + 8*(l>>4), N = l&15
    const float bias0 = bias[lo];
    const float lb0   = leftb[lo];
    const float rb0   = rightb[lo];
    const float bias1 = valid1 ? bias[n1c]   : 0.0f;
    const float lb1   = valid1 ? leftb[n1c]  : 0.0f;
    const float rb1   = valid1 ? rightb[n1c] : 0.0f;

#pragma unroll
    for (int r = 0; r < 8; ++r) {
        const int m    = r + 8 * hi;
        const int grow = row0 + m;
        const int t    = grow & (T_LEN - 1);
        {
            float v = c0[r] + bias0;
            if (t == 0)         v += lb0;
            if (t == T_LEN - 1) v += rb0;
            pot[(size_t)grow * N_TAGS + lo] = v;
        }
        if (valid1) {
            float v = c1[r] + bias1;
            if (t == 0)         v += lb1;
            if (t == T_LEN - 1) v += rb1;
            pot[(size_t)grow * N_TAGS + n1] = v;
        }
    }
}

// ---------------------------------------------------------------------------
// Kernel 2: Viterbi forward scan + backtrace. One wave32 per batch element.
// Entire pot[b] slice (96KB) and backpointers (24KB) staged in LDS so the
// 1024-step serial recurrence only touches low-latency LDS.
// ---------------------------------------------------------------------------
__global__ __launch_bounds__(32) void crf_viterbi(
    const float* __restrict__ pot,    // (B, T, N)
    const float* __restrict__ trans,  // (N, N)
    float* __restrict__ decoded)      // (B, T) stored as f32
{
    __shared__ __align__(16) float         potS[T_LEN * N_TAGS];       // 96 KB
    __shared__ __align__(16) float         alphaS[N_TAGS + 8];
    __shared__               unsigned char bpS[(T_LEN - 1) * N_TAGS];  // ~24 KB

    const int b    = blockIdx.x;
    const int lane = threadIdx.x;
    const bool active = (lane < N_TAGS);

    // Bulk-stage this batch's pot slice (6144 float4s) directly into LDS
    // via the CDNA5 async-LDS path (no VGPR round trip).
    stage_to_lds(potS, pot + (size_t)b * T_LEN * N_TAGS,
                 (T_LEN * N_TAGS) / 4, lane, 32);

    // Lane n owns tag n: keep trans[:, n] resident in 24 VGPRs.
    float transCol[N_TAGS];
    if (active) {
#pragma unroll
        for (int m = 0; m < N_TAGS; ++m)
            transCol[m] = trans[m * N_TAGS + lane];
    }
    __syncthreads();

    if (active)
        alphaS[lane] = potS[lane];             // alpha_0 = pot[:,0,:] (incl. left bnd)
    __syncthreads();

    for (int t = 1; t < T_LEN; ++t) {
        // Broadcast-read alpha as 6 x b128 (all lanes same address -> no conflicts)
        float av[N_TAGS];
#pragma unroll
        for (int q = 0; q < N_TAGS / 4; ++q) {
            float4 w = ((const float4*)alphaS)[q];
            av[4 * q + 0] = w.x;
            av[4 * q + 1] = w.y;
            av[4 * q + 2] = w.z;
            av[4 * q + 3] = w.w;
        }
        float best = -INFINITY;
        int   arg  = 0;
#pragma unroll
        for (int m = 0; m < N_TAGS; ++m) {
            float v = av[m] + transCol[m];
            if (v > best) { best = v; arg = m; }   // strict '>' == first-max (argmax)
        }
        __syncthreads();
        if (active) {
            alphaS[lane] = best + potS[t * N_TAGS + lane];
            bpS[(t - 1) * N_TAGS + lane] = (unsigned char)arg;
        }
        __syncthreads();
    }

    // Final argmax + backtrace (single lane; ~1K serial LDS byte loads).
    if (lane == 0) {
        int   tag  = 0;
        float best = alphaS[0];
#pragma unroll
        for (int m = 1; m < N_TAGS; ++m) {
            if (alphaS[m] > best) { best = alphaS[m]; tag = m; }
        }
        float* drow = decoded + (size_t)b * T_LEN;
        drow[T_LEN - 1] = (float)tag;
        for (int t = T_LEN - 2; t >= 0; --t) {
            tag = (int)bpS[t * N_TAGS + tag];
            drow[t] = (float)tag;
        }
    }
}

// ---------------------------------------------------------------------------
// Kernel 3: tuple tail outputs: lens (all T) and trans passthrough.
// ---------------------------------------------------------------------------
__global__ void crf_tail(const float* __restrict__ trans,
                         float* __restrict__ out_lens,
                         float* __restrict__ out_trans)
{
    const int i = blockIdx.x * blockDim.x + threadIdx.x;
    if (i < B_SZ)            out_lens[i]  = (float)T_LEN;
    if (i < N_TAGS * N_TAGS) out_trans[i] = trans[i];
}

// ---------------------------------------------------------------------------
extern "C" void kernel_launch(void* const* d_in, const int* in_sizes, int n_in,
                              void* d_out, int out_size, void* d_ws, size_t ws_size,
                              hipStream_t stream)
{
    const float* x     = (const float*)d_in[0];   // (B, T, D)
    const float* W     = (const float*)d_in[1];   // (D, N)
    const float* bias  = (const float*)d_in[2];   // (N)
    const float* trans = (const float*)d_in[3];   // (N, N)
    const float* lb    = (const float*)d_in[4];   // (N)
    const float* rb    = (const float*)d_in[5];   // (N)

    float* out     = (float*)d_out;
    float* decoded = out;                                   // B*T
    float* pot     = decoded + (size_t)B_SZ * T_LEN;        // B*T*N
    float* lens    = pot + (size_t)B_SZ * T_LEN * N_TAGS;   // B
    float* otrans  = lens + B_SZ;                           // N*N

    // Phase 1: 4096 M-tiles of 16 rows, 8 waves/block.
    crf_pot_gemm<<<512, 256, 0, stream>>>(x, W, bias, lb, rb, pot);
    // Phase 2: one wave32 workgroup per batch element.
    crf_viterbi<<<B_SZ, 32, 0, stream>>>(pot, trans, decoded);
    // Phase 3: tail outputs.
    crf_tail<<<3, 256, 0, stream>>>(trans, lens, otrans);
}